// LightGCN_9955734192814
// MI455X (gfx1250) — compile-verified
//
#include <hip/hip_runtime.h>

typedef float v2f __attribute__((ext_vector_type(2)));
typedef float v8f __attribute__((ext_vector_type(8)));

#define ALPHA 0.25f   // 1/(NUM_LAYERS+1), NUM_LAYERS=3

// ---------------- elementwise helpers ----------------

__global__ __launch_bounds__(256) void zero_f32_kernel(float* __restrict__ p, long long n) {
    long long i = (long long)blockIdx.x * blockDim.x + threadIdx.x;
    if (i < n) p[i] = 0.0f;
}

__global__ __launch_bounds__(256) void deg_kernel(const int* __restrict__ dst,
                                                  float* __restrict__ deg, int E) {
    int e = blockIdx.x * blockDim.x + threadIdx.x;
    if (e < E) unsafeAtomicAdd(&deg[dst[e]], 1.0f);
}

__global__ __launch_bounds__(256) void dis_kernel(float* __restrict__ deg, int N) {
    int i = blockIdx.x * blockDim.x + threadIdx.x;
    if (i < N) {
        float d = deg[i];
        deg[i] = (d > 0.0f) ? rsqrtf(d) : 0.0f;   // deg>=1 when >0, matches ref
    }
}

__global__ __launch_bounds__(256) void norm_kernel(const int* __restrict__ src,
                                                   const int* __restrict__ dst,
                                                   const float* __restrict__ dis,
                                                   float* __restrict__ norm, int E) {
    int e = blockIdx.x * blockDim.x + threadIdx.x;
    if (e < E) norm[e] = dis[src[e]] * dis[dst[e]];
}

// x_cur = emb ; out_acc = alpha * emb
__global__ __launch_bounds__(256) void init_kernel(const float* __restrict__ emb,
                                                   float* __restrict__ x_cur,
                                                   float* __restrict__ out_acc,
                                                   long long n) {
    long long i = (long long)blockIdx.x * blockDim.x + threadIdx.x;
    if (i < n) {
        float v = emb[i];
        x_cur[i] = v;
        out_acc[i] = v * ALPHA;
    }
}

// out_acc += alpha * x
__global__ __launch_bounds__(256) void accum_kernel(const float* __restrict__ x,
                                                    float* __restrict__ out_acc,
                                                    long long n) {
    long long i = (long long)blockIdx.x * blockDim.x + threadIdx.x;
    if (i < n) out_acc[i] += ALPHA * x[i];
}

// ---------------- SpMM: one wave32 per edge, 2 dims per lane ----------------
// x_next[dst, :] += norm * x_cur[src, :]   (EMB_DIM = 64)
__global__ __launch_bounds__(256) void spmm_kernel(const int* __restrict__ src,
                                                   const int* __restrict__ dst,
                                                   const float* __restrict__ norm,
                                                   const float* __restrict__ x_cur,
                                                   float* __restrict__ x_next,
                                                   int E) {
    long long tid = (long long)blockIdx.x * blockDim.x + threadIdx.x;
    long long e   = tid >> 5;                 // wave-uniform edge id
    if (e >= E) return;                       // wave-uniform exit
    int d2 = ((int)tid & 31) * 2;             // lane's pair of dims

    int   s = src[e];                         // wave-uniform loads (broadcast)
    int   t = dst[e];
    float w = norm[e];

    const float2* xp = (const float2*)(x_cur + (long long)s * 64 + d2);
    float2 v = *xp;
    float* op = x_next + (long long)t * 64 + d2;
    unsafeAtomicAdd(op,     w * v.x);         // global_atomic_add_f32 (L2)
    unsafeAtomicAdd(op + 1, w * v.y);
}

// ---------------- scoring: 16 query pairs per wave via V_WMMA_F32_16X16X4_F32 ----
// D = A(16x64) * B(64x16) accumulated in 16 K-chunks of 4; diag(D)_i = dot(a_i,b_i).
// f32 A 16x4 fragment: VGPR0 = K={0 | 2}, VGPR1 = K={1 | 3} for lanes {0-15 | 16-31},
// row M = lane%16  ->  each lane's fragment is a contiguous float2 at
//                      out[row*64 + k0 + (lane>=16 ? 2 : 0)].  B mirrors with N<->M.
__global__ __launch_bounds__(256) void score_kernel(const float* __restrict__ out,
                                                    const int* __restrict__ qa,
                                                    const int* __restrict__ qb,
                                                    float* __restrict__ res, int Q) {
    long long tid  = (long long)blockIdx.x * blockDim.x + threadIdx.x;
    int       lane = threadIdx.x & 31;
    long long wave = tid >> 5;
    int qbase = (int)(wave * 16);
    if (qbase >= Q) return;                   // wave-uniform exit, EXEC stays full

    int m = lane & 15;
    int q = qbase + m;
    if (q >= Q) q = Q - 1;                    // clamp (no EXEC change)

    long long ra = (long long)qa[q] * 64;
    long long rb = (long long)qb[q] * 64;
    int koff = (lane >= 16) ? 2 : 0;
    const float* pa = out + ra + koff;
    const float* pb = out + rb + koff;

    v8f acc = {};
#pragma unroll
    for (int k0 = 0; k0 < 64; k0 += 4) {
        v2f a = *(const v2f*)(pa + k0);
        v2f b = *(const v2f*)(pb + k0);
        // 8 args: (neg_a, A, neg_b, B, c_mod, C, reuse_a, reuse_b)
        acc = __builtin_amdgcn_wmma_f32_16x16x4_f32(
            false, a, false, b, (short)0, acc, false, false);
    }

    // diag element i: i<8 -> VGPR i @ lane i ; i>=8 -> VGPR i-8 @ lane i+16
    int r = lane & 7;
    float val = acc[0];
    switch (r) {
        case 1: val = acc[1]; break;
        case 2: val = acc[2]; break;
        case 3: val = acc[3]; break;
        case 4: val = acc[4]; break;
        case 5: val = acc[5]; break;
        case 6: val = acc[6]; break;
        case 7: val = acc[7]; break;
        default: break;
    }
    bool writer = (lane < 8) || (lane >= 24);
    int  qi     = qbase + ((lane < 8) ? lane : (lane - 16));
    if (writer && qi < Q) res[qi] = val;
}

// ---------------- host ----------------

static inline size_t align_elems(size_t n) { return (n + 63) & ~(size_t)63; }

extern "C" void kernel_launch(void* const* d_in, const int* in_sizes, int n_in,
                              void* d_out, int out_size, void* d_ws, size_t ws_size,
                              hipStream_t stream) {
    const float* emb = (const float*)d_in[0];
    const int*   ei  = (const int*)d_in[1];   // edge_index     [2, E] flat
    const int*   eli = (const int*)d_in[2];   // edge_label_idx [2, Q] flat

    const int N = in_sizes[0] / 64;           // 100000
    const int E = in_sizes[1] / 2;            // 3200000
    const int Q = in_sizes[2] / 2;            // 1000000

    const int* src = ei;
    const int* dst = ei + E;
    const int* qa  = eli;
    const int* qb  = eli + Q;

    const long long NF = (long long)N * 64;   // floats per table

    // workspace carve-up (all 256B aligned)
    float* deg   = (float*)d_ws;                       // N  (becomes dis in place)
    float* norm  = deg  + align_elems((size_t)N);      // E
    float* x0    = norm + align_elems((size_t)E);      // N*64
    float* x1    = x0   + align_elems((size_t)NF);     // N*64
    float* oacc  = x1   + align_elems((size_t)NF);     // N*64

    const int B = 256;
    dim3 blk(B);

    // 1) degree
    zero_f32_kernel<<<dim3((unsigned)((N + B - 1) / B)), blk, 0, stream>>>(deg, N);
    deg_kernel<<<dim3((unsigned)((E + B - 1) / B)), blk, 0, stream>>>(dst, deg, E);
    dis_kernel<<<dim3((unsigned)((N + B - 1) / B)), blk, 0, stream>>>(deg, N);

    // 2) per-edge norm
    norm_kernel<<<dim3((unsigned)((E + B - 1) / B)), blk, 0, stream>>>(src, dst, deg, norm, E);

    // 3) init x_cur, out_acc
    init_kernel<<<dim3((unsigned)((NF + B - 1) / B)), blk, 0, stream>>>(emb, x0, oacc, NF);

    // 4) 3 propagation layers
    float* cur = x0;
    float* nxt = x1;
    const long long spmm_threads = (long long)E * 32;
    for (int layer = 0; layer < 3; ++layer) {
        zero_f32_kernel<<<dim3((unsigned)((NF + B - 1) / B)), blk, 0, stream>>>(nxt, NF);
        spmm_kernel<<<dim3((unsigned)((spmm_threads + B - 1) / B)), blk, 0, stream>>>(
            src, dst, norm, cur, nxt, E);
        accum_kernel<<<dim3((unsigned)((NF + B - 1) / B)), blk, 0, stream>>>(nxt, oacc, NF);
        float* t = cur; cur = nxt; nxt = t;
    }

    // 5) WMMA dot-product scoring
    long long waves = ((long long)Q + 15) / 16;
    long long sthreads = waves * 32;
    score_kernel<<<dim3((unsigned)((sthreads + B - 1) / B)), blk, 0, stream>>>(
        oacc, qa, qb, (float*)d_out, Q);
}